// GSAGE_3layer_71949292143006
// MI455X (gfx1250) — compile-verified
//
#include <hip/hip_runtime.h>
#include <hip/hip_bf16.h>

#define N_NODES 100000
#define E_EDGES 1600000
#define D_HID   64
#define D_OUT_F 10

typedef float v2f __attribute__((ext_vector_type(2)));
typedef float v8f __attribute__((ext_vector_type(8)));

// ---------------------------------------------------------------------------
// Degree count: cnt[dst] += 1 over all edges (edge_index row-major [2,E])
// ---------------------------------------------------------------------------
__global__ void deg_kernel(const int* __restrict__ ei, float* __restrict__ cnt) {
    int g = blockIdx.x * blockDim.x + threadIdx.x;
    if (g < E_EDGES) {
        int dst = ei[E_EDGES + g];
        unsafeAtomicAdd(&cnt[dst], 1.0f);
    }
}

// recip[i] = 1 / max(cnt[i], 1)
__global__ void recip_kernel(float* __restrict__ r) {
    int g = blockIdx.x * blockDim.x + threadIdx.x;
    if (g < N_NODES) r[g] = 1.0f / fmaxf(r[g], 1.0f);
}

// ---------------------------------------------------------------------------
// Edge-parallel scatter-add: agg[dst, f] += xin[src, f], D=64 features
// thread = (edge, feature) pair; 64 consecutive threads share one edge ->
// coalesced row read of xin[src], contiguous atomic region at agg[dst].
// ---------------------------------------------------------------------------
__global__ void scatter_kernel(const int* __restrict__ ei,
                               const float* __restrict__ xin,
                               float* __restrict__ agg) {
    long long g = (long long)blockIdx.x * blockDim.x + threadIdx.x;
    if (g < (long long)E_EDGES * 64) {
        int e = (int)(g >> 6);
        int f = (int)(g & 63);
        int s = ei[e];
        int d = ei[E_EDGES + e];
        unsafeAtomicAdd(&agg[(size_t)d * 64 + f], xin[(size_t)s * 64 + f]);
    }
}

// ---------------------------------------------------------------------------
// Fused SAGE GEMM: out = (agg*recip) @ Wl^T + xin @ Wr^T + b, optional row-L2
// One wave32 per 16-row node tile. WMMA F32 16x16x4, K=64 (16 k-steps),
// 2 WMMAs per k-step (mean path + self path) into shared accumulators.
//
// A layout (16x4 f32): lane m=l&15 holds row m; kh=l>>4; v0=K(2kh), v1=K(2kh+1)
// B layout (4x16 f32): lane n=l&15 holds col n; v0=row(2kh), v1=row(2kh+1)
// C/D layout (16x16 f32): element r of lane l = D[row r+8*kh, col l&15]
// ---------------------------------------------------------------------------
template <int DOUT, bool NORM>
__global__ void sage_gemm_kernel(const float* __restrict__ agg,
                                 const float* __restrict__ recip,
                                 const float* __restrict__ xin,
                                 const float* __restrict__ Wl,
                                 const float* __restrict__ Wr,
                                 const float* __restrict__ bias,
                                 float* __restrict__ out) {
    constexpr int NTILES = (DOUT + 15) / 16;
    const int lane = threadIdx.x & 31;
    const int wave = threadIdx.x >> 5;
    const int tile = blockIdx.x * (blockDim.x >> 5) + wave;
    const int row0 = tile * 16;
    if (row0 >= N_NODES) return;  // wave-uniform: EXEC stays all-ones

    const int m  = lane & 15;
    const int kh = lane >> 4;
    const int row = row0 + m;

    const float rcp = recip[row];
    const float* aggp = agg + (size_t)row * 64 + kh * 2;
    const float* xp   = xin + (size_t)row * 64 + kh * 2;

    // init accumulators with bias (same per column)
    v8f acc[NTILES];
    for (int nt = 0; nt < NTILES; ++nt) {
        int n = nt * 16 + m;
        float bv = (n < DOUT) ? bias[n] : 0.0f;
        for (int r = 0; r < 8; ++r) acc[nt][r] = bv;
    }

    for (int ks = 0; ks < 16; ++ks) {
        v2f am, ax;
        am.x = aggp[ks * 4 + 0] * rcp;
        am.y = aggp[ks * 4 + 1] * rcp;
        ax.x = xp[ks * 4 + 0];
        ax.y = xp[ks * 4 + 1];
        for (int nt = 0; nt < NTILES; ++nt) {
            int n = nt * 16 + m;
            v2f bl = {0.0f, 0.0f}, br = {0.0f, 0.0f};
            if (n < DOUT) {  // zero-pad columns beyond DOUT
                const float* wlp = Wl + (size_t)n * 64 + ks * 4 + kh * 2;
                const float* wrp = Wr + (size_t)n * 64 + ks * 4 + kh * 2;
                bl.x = wlp[0]; bl.y = wlp[1];
                br.x = wrp[0]; br.y = wrp[1];
            }
            acc[nt] = __builtin_amdgcn_wmma_f32_16x16x4_f32(
                false, am, false, bl, (short)0, acc[nt], false, false);
            acc[nt] = __builtin_amdgcn_wmma_f32_16x16x4_f32(
                false, ax, false, br, (short)0, acc[nt], false, false);
        }
    }

    if (NORM) {
        // per-row L2 norm: reduce squares across the 16 lanes of each kh-group
        for (int r = 0; r < 8; ++r) {
            float ss = 0.0f;
            for (int nt = 0; nt < NTILES; ++nt) ss += acc[nt][r] * acc[nt][r];
            ss += __shfl_xor(ss, 1, 32);
            ss += __shfl_xor(ss, 2, 32);
            ss += __shfl_xor(ss, 4, 32);
            ss += __shfl_xor(ss, 8, 32);
            float inv = 1.0f / fmaxf(sqrtf(ss), 1e-12f);
            for (int nt = 0; nt < NTILES; ++nt) acc[nt][r] *= inv;
        }
    }

    for (int nt = 0; nt < NTILES; ++nt) {
        int n = nt * 16 + m;
        if (n < DOUT) {
            for (int r = 0; r < 8; ++r) {
                int orow = row0 + r + 8 * kh;
                out[(size_t)orow * DOUT + n] = acc[nt][r];
            }
        }
    }
}

// ---------------------------------------------------------------------------
// BatchNorm pass 1: per-column sum / sumsq (partial per-thread, then atomics)
// ---------------------------------------------------------------------------
__global__ void bnsum_kernel(const float* __restrict__ h,
                             float* __restrict__ bsum, float* __restrict__ bsq) {
    int g = blockIdx.x * blockDim.x + threadIdx.x;
    int c = g & 63;
    int rstride = (gridDim.x * blockDim.x) >> 6;
    float s = 0.0f, q = 0.0f;
    for (int row = g >> 6; row < N_NODES; row += rstride) {
        float v = h[(size_t)row * 64 + c];
        s += v;
        q += v * v;
    }
    unsafeAtomicAdd(&bsum[c], s);
    unsafeAtomicAdd(&bsq[c], q);
}

// BatchNorm pass 2: fold into per-column scale/shift
__global__ void bnfinal_kernel(const float* __restrict__ bsum,
                               const float* __restrict__ bsq,
                               const float* __restrict__ gamma,
                               const float* __restrict__ beta,
                               float* __restrict__ sc, float* __restrict__ sh) {
    int c = threadIdx.x;
    if (c < 64) {
        const float invN = 1.0f / (float)N_NODES;
        float mu  = bsum[c] * invN;
        float var = bsq[c] * invN - mu * mu;
        float s   = gamma[c] * rsqrtf(var + 1e-5f);
        sc[c] = s;
        sh[c] = beta[c] - mu * s;
    }
}

// BatchNorm apply + ReLU, in place
__global__ void bnapply_kernel(float* __restrict__ h,
                               const float* __restrict__ sc,
                               const float* __restrict__ sh) {
    size_t g = (size_t)blockIdx.x * blockDim.x + threadIdx.x;
    if (g < (size_t)N_NODES * 64) {
        int c = (int)(g & 63);
        float v = h[g] * sc[c] + sh[c];
        h[g] = v > 0.0f ? v : 0.0f;
    }
}

// ---------------------------------------------------------------------------
extern "C" void kernel_launch(void* const* d_in, const int* in_sizes, int n_in,
                              void* d_out, int out_size, void* d_ws, size_t ws_size,
                              hipStream_t stream) {
    const float* x   = (const float*)d_in[0];
    const int*   ei  = (const int*)  d_in[1];
    const float* Wl1 = (const float*)d_in[2];
    const float* Wr1 = (const float*)d_in[3];
    const float* b1  = (const float*)d_in[4];
    const float* g1  = (const float*)d_in[5];
    const float* be1 = (const float*)d_in[6];
    const float* Wl2 = (const float*)d_in[7];
    const float* Wr2 = (const float*)d_in[8];
    const float* b2  = (const float*)d_in[9];
    const float* g2  = (const float*)d_in[10];
    const float* be2 = (const float*)d_in[11];
    const float* Wl3 = (const float*)d_in[12];
    const float* Wr3 = (const float*)d_in[13];
    const float* b3  = (const float*)d_in[14];
    float* out = (float*)d_out;

    // workspace carve-up (floats)
    float* ws    = (float*)d_ws;
    float* agg   = ws;                            // N*64
    float* recip = agg + (size_t)N_NODES * 64;    // N
    float* h1    = recip + N_NODES;               // N*64
    float* h2    = h1 + (size_t)N_NODES * 64;     // N*64
    float* bns   = h2 + (size_t)N_NODES * 64;     // 64
    float* bnq   = bns + 64;                      // 64
    float* bnsc  = bnq + 64;                      // 64
    float* bnsh  = bnsc + 64;                     // 64

    const int TPB = 256;
    const int degBlocks  = (E_EDGES + TPB - 1) / TPB;
    const int nBlocks    = (N_NODES + TPB - 1) / TPB;
    const long long scatterThreads = (long long)E_EDGES * 64;
    const int scatBlocks = (int)((scatterThreads + TPB - 1) / TPB);
    const int tiles      = (N_NODES + 15) / 16;
    const int gemmBlocks = (tiles + 7) / 8;  // 8 waves (wave32) per block
    const size_t elemsND = (size_t)N_NODES * 64;
    const int ewBlocks   = (int)((elemsND + TPB - 1) / TPB);

    // degrees (shared by all 3 layers)
    hipMemsetAsync(recip, 0, N_NODES * sizeof(float), stream);
    deg_kernel<<<degBlocks, TPB, 0, stream>>>(ei, recip);
    recip_kernel<<<nBlocks, TPB, 0, stream>>>(recip);

    // ---- layer 1 ----
    hipMemsetAsync(agg, 0, elemsND * sizeof(float), stream);
    scatter_kernel<<<scatBlocks, TPB, 0, stream>>>(ei, x, agg);
    sage_gemm_kernel<64, true><<<gemmBlocks, TPB, 0, stream>>>(
        agg, recip, x, Wl1, Wr1, b1, h1);
    hipMemsetAsync(bns, 0, 128 * sizeof(float), stream);
    bnsum_kernel<<<512, TPB, 0, stream>>>(h1, bns, bnq);
    bnfinal_kernel<<<1, 64, 0, stream>>>(bns, bnq, g1, be1, bnsc, bnsh);
    bnapply_kernel<<<ewBlocks, TPB, 0, stream>>>(h1, bnsc, bnsh);

    // ---- layer 2 ----
    hipMemsetAsync(agg, 0, elemsND * sizeof(float), stream);
    scatter_kernel<<<scatBlocks, TPB, 0, stream>>>(ei, h1, agg);
    sage_gemm_kernel<64, true><<<gemmBlocks, TPB, 0, stream>>>(
        agg, recip, h1, Wl2, Wr2, b2, h2);
    hipMemsetAsync(bns, 0, 128 * sizeof(float), stream);
    bnsum_kernel<<<512, TPB, 0, stream>>>(h2, bns, bnq);
    bnfinal_kernel<<<1, 64, 0, stream>>>(bns, bnq, g2, be2, bnsc, bnsh);
    bnapply_kernel<<<ewBlocks, TPB, 0, stream>>>(h2, bnsc, bnsh);

    // ---- layer 3 (D_OUT=10, no normalize) -> d_out ----
    hipMemsetAsync(agg, 0, elemsND * sizeof(float), stream);
    scatter_kernel<<<scatBlocks, TPB, 0, stream>>>(ei, h2, agg);
    sage_gemm_kernel<D_OUT_F, false><<<gemmBlocks, TPB, 0, stream>>>(
        agg, recip, h2, Wl3, Wr3, b3, out);
}